// CosineLoss_45689862095127
// MI455X (gfx1250) — compile-verified
//
#include <hip/hip_runtime.h>

// ---- problem constants (from reference) ----
#define B_IMG   512
#define P_PATCH 45
#define D_DIM   2048
#define C_CLS   4
#define THRESH  0.6f
#define EPSV    1e-8f

// LDS row stride for avg copy: 2048 + 16 words so the 4 live B-rows
// (n*stride) land in distinct banks ((n*16)%64 = 0,16,32,48).
#define AVG_STRIDE 2064
#define NTHREADS   96   // 3 wave32s: one 16-patch WMMA tile per wave
#define UN         8    // k-steps per pipelined block (32 K-elements)
#define KB         (UN * 4)

typedef __attribute__((ext_vector_type(2))) float v2f;
typedef __attribute__((ext_vector_type(8))) float v8f;

__global__ __launch_bounds__(NTHREADS)
void cosine_loss_stage1(const float* __restrict__ features,       // [B,P,D]
                        const float* __restrict__ avg,            // [C,D]
                        const float* __restrict__ outputs,        // [B,P,C]
                        const float* __restrict__ labels_onehot,  // [B,C]
                        const float* __restrict__ weights,        // [C]
                        float* __restrict__ block_partial)        // [B]
{
    __shared__ float ldsAvg[C_CLS * AVG_STRIDE];   // avg, bank-skewed
    __shared__ float ldsS[3][16][16];              // WMMA D tiles
    __shared__ float ldsN[3 * 32];                 // per-lane sumsq partials
    __shared__ float ldsPart[NTHREADS * C_CLS];    // proto-norm partials
    __shared__ float ldsCN[C_CLS];                 // max(||avg_c||, eps)
    __shared__ int   ldsLabel;
    __shared__ float ldsLoss[P_PATCH];

    const int b    = blockIdx.x;
    const int tid  = threadIdx.x;
    const int wave = tid >> 5;          // patch tile id 0..2
    const int lane = tid & 31;

    // ---- preload prototypes into LDS + per-class sum-of-squares (fixed order) ----
    float part[C_CLS] = {0.f, 0.f, 0.f, 0.f};
    for (int idx = tid; idx < C_CLS * D_DIM; idx += NTHREADS) {
        int c = idx >> 11;              // D_DIM == 2048
        int k = idx & (D_DIM - 1);
        float v = avg[idx];
        ldsAvg[c * AVG_STRIDE + k] = v;
        part[c] += v * v;
    }
#pragma unroll
    for (int c = 0; c < C_CLS; ++c) ldsPart[tid * C_CLS + c] = part[c];

    if (tid == 0) {  // argmax of exact one-hot row -> label index
        const float* lo = labels_onehot + (size_t)b * C_CLS;
        int li = 0; float best = lo[0];
        for (int c = 1; c < C_CLS; ++c) { if (lo[c] > best) { best = lo[c]; li = c; } }
        ldsLabel = li;
    }
    __syncthreads();
    if (tid < C_CLS) {
        float s = 0.f;
        for (int t = 0; t < NTHREADS; ++t) s += ldsPart[t * C_CLS + tid];
        ldsCN[tid] = fmaxf(sqrtf(s), EPSV);
    }
    __syncthreads();

    // ---- fp32 WMMA K-loop: S_tile(16x16) += A(16x4) x B(4x16) ----
    // A frag (16x4 f32): lane L -> A[L][k],A[L][k+1]; lane L+16 -> A[L][k+2..3]
    // B frag (4x16 f32): lane n -> B[k][n],B[k+1][n]; lane n+16 -> B[k+2..3][n]
    // Branchless operand fetch:
    //   - A rows clamped to 44: duplicated rows only feed D rows we never read.
    //   - B cols n>=4 read LDS row 3 (same-address broadcast) then *0 mask.
    const int  row   = wave * 16 + (lane & 15);
    const int  crow  = (row < P_PATCH) ? row : (P_PATCH - 1);
    const int  koff  = (lane >> 4) * 2;
    const int  nidx  = lane & 15;
    const int  bn    = (nidx < C_CLS) ? nidx : (C_CLS - 1);
    const float mask = (nidx < C_CLS) ? 1.f : 0.f;
    const v2f  bmask = {mask, mask};
    const float* frow = features + ((size_t)b * P_PATCH + crow) * D_DIM + koff;
    const float* brow = ldsAvg + bn * AVG_STRIDE + koff;

    v8f   acc = {};
    float ns0 = 0.f, ns1 = 0.f;
    v2f aN[UN], bN[UN];

    // prologue: block 0 in flight
#pragma unroll
    for (int i = 0; i < UN; ++i) {
        aN[i] = __builtin_nontemporal_load((const v2f*)(frow + i * 4));
        bN[i] = *(const v2f*)(brow + i * 4);
    }
    // steady state: issue block k's loads, then compute block k-1
    for (int k = KB; k < D_DIM; k += KB) {
        v2f aC[UN], bC[UN];
#pragma unroll
        for (int i = 0; i < UN; ++i) { aC[i] = aN[i]; bC[i] = bN[i]; }
#pragma unroll
        for (int i = 0; i < UN; ++i) {
            aN[i] = __builtin_nontemporal_load((const v2f*)(frow + k + i * 4));
            bN[i] = *(const v2f*)(brow + k + i * 4);
        }
#pragma unroll
        for (int i = 0; i < UN; ++i) {
            v2f bb = bC[i] * bmask;                    // v_pk_mul_f32, no EXEC games
            ns0 = fmaf(aC[i].x, aC[i].x, ns0);         // two independent fma chains
            ns1 = fmaf(aC[i].y, aC[i].y, ns1);
            acc = __builtin_amdgcn_wmma_f32_16x16x4_f32(
                      false, aC[i], false, bb, (short)0, acc, false, false);
        }
    }
    // epilogue: last block
#pragma unroll
    for (int i = 0; i < UN; ++i) {
        v2f bb = bN[i] * bmask;
        ns0 = fmaf(aN[i].x, aN[i].x, ns0);
        ns1 = fmaf(aN[i].y, aN[i].y, ns1);
        acc = __builtin_amdgcn_wmma_f32_16x16x4_f32(
                  false, aN[i], false, bb, (short)0, acc, false, false);
    }
    const float nsq = ns0 + ns1;

    // spill D tile: VGPR r -> D[M=r][N=lane] (lanes 0-15), D[M=r+8][N=lane-16]
    {
        const int mbase = (lane < 16) ? 0 : 8;
        const int n = lane & 15;
#pragma unroll
        for (int r = 0; r < 8; ++r) ldsS[wave][mbase + r][n] = acc[r];
        ldsN[wave * 32 + lane] = nsq;
    }
    __syncthreads();

    // ---- per-patch epilogue: cosine sims -> pseudo target -> weighted CE ----
    if (tid < P_PATCH) {
        const int p = tid, t = p >> 4, r = p & 15;
        float n2 = ldsN[t * 32 + r] + ldsN[t * 32 + r + 16];
        float fn = fmaxf(sqrtf(n2), EPSV);
        int   li = ldsLabel;
        float simB = ldsS[t][r][0]  / (fn * ldsCN[0]);
        float simS = ldsS[t][r][li] / (fn * ldsCN[li]);
        bool keep   = (simB > simS) && (simB > THRESH);
        bool pseudo = (li > 0) && !keep;
        int  tc = pseudo ? li : 0;   // target is one-hot at tc either way

        const float* o = outputs + ((size_t)b * P_PATCH + p) * C_CLS;
        float o0 = o[0], o1 = o[1], o2 = o[2], o3 = o[3];
        // softmax(outputs)
        float m  = fmaxf(fmaxf(o0, o1), fmaxf(o2, o3));
        float e0 = expf(o0 - m), e1 = expf(o1 - m), e2 = expf(o2 - m), e3 = expf(o3 - m);
        float inv = 1.f / (e0 + e1 + e2 + e3);
        float s0 = e0 * inv, s1 = e1 * inv, s2 = e2 * inv, s3 = e3 * inv;
        // log_softmax of the (already) softmaxed values (faithful double softmax)
        float lse = logf(expf(s0) + expf(s1) + expf(s2) + expf(s3));
        float st  = (tc == 0) ? s0 : (tc == 1) ? s1 : (tc == 2) ? s2 : s3;
        ldsLoss[p] = -weights[tc] * (st - lse);
    }
    __syncthreads();
    if (tid == 0) {  // fixed-order per-image sum (deterministic)
        float s = 0.f;
        for (int p = 0; p < P_PATCH; ++p) s += ldsLoss[p];
        block_partial[b] = s;
    }
}

__global__ __launch_bounds__(256)
void cosine_loss_stage2(const float* __restrict__ part, float* __restrict__ out)
{
    __shared__ float l[256];
    const int tid = threadIdx.x;
    l[tid] = part[tid] + part[tid + 256];
    __syncthreads();
#pragma unroll
    for (int off = 128; off > 0; off >>= 1) {
        if (tid < off) l[tid] += l[tid + off];
        __syncthreads();
    }
    if (tid == 0) out[0] = l[0] * (1.0f / (float)(B_IMG * P_PATCH));
}

extern "C" void kernel_launch(void* const* d_in, const int* in_sizes, int n_in,
                              void* d_out, int out_size, void* d_ws, size_t ws_size,
                              hipStream_t stream)
{
    const float* features = (const float*)d_in[0];  // [512,45,2048] f32
    const float* avg      = (const float*)d_in[1];  // [4,2048]      f32
    const float* outputs  = (const float*)d_in[2];  // [512,45,4]    f32
    const float* labels   = (const float*)d_in[3];  // [512,4]       f32
    const float* weights  = (const float*)d_in[4];  // [4]           f32
    float* partials = (float*)d_ws;                 // 512 floats of scratch
    float* out      = (float*)d_out;                // scalar f32

    cosine_loss_stage1<<<B_IMG, NTHREADS, 0, stream>>>(
        features, avg, outputs, labels, weights, partials);
    cosine_loss_stage2<<<1, 256, 0, stream>>>(partials, out);
}